// LifSpike_83416854823334
// MI455X (gfx1250) — compile-verified
//
#include <hip/hip_runtime.h>
#include <cstdint>
#include <cstddef>

// LIF scan: mem = BETA*mem + x[t]; spk = mem>=THRESH; mem = fired ? 0 : mem.
// Shape [T=8, B=32, C=128, H=32, W=32] f32. Pure streaming: 268 MB total
// traffic -> ~11.5us floor at 23.3 TB/s. Data path: async global->LDS b128
// (ASYNCcnt), ds_load_b128, nontemporal b128 stores.

#define THRESH 0.5f
#define BETA 0.25f

typedef float v4f __attribute__((ext_vector_type(4)));
typedef int   v4i __attribute__((vector_size(16)));   // matches builtin's V4i

constexpr int T_STEPS = 8;
constexpr int VEC = 4;                  // floats per thread per timestep
constexpr int BLOCK = 256;              // 8 wave32s
constexpr int CHUNK = BLOCK * VEC;      // 1024 floats per timestep per block

// ---- CDNA5 async global->LDS copy (ASYNCcnt-tracked) -----------------------

__device__ __forceinline__ void async_load_b128(const float* gsrc, void* lds_dst) {
#if defined(__HIP_DEVICE_COMPILE__) && __has_builtin(__builtin_amdgcn_global_load_async_to_lds_b128)
    // Signature (from hipcc diagnostic): (AS1 v4i*, AS3 v4i*, Imm offset, Imm cpol)
    __builtin_amdgcn_global_load_async_to_lds_b128(
        (__attribute__((address_space(1))) v4i*)(uintptr_t)gsrc,
        (__attribute__((address_space(3))) v4i*)(uint32_t)(uintptr_t)lds_dst,
        0, 0);
#elif defined(__HIP_DEVICE_COMPILE__)
    // Fallback: raw instruction. Low 32 bits of the flat address of an LDS
    // object are the LDS byte offset (ISA 10.2).
    uint32_t lds_off = (uint32_t)(uintptr_t)lds_dst;
    asm volatile("global_load_async_to_lds_b128 %0, %1, off"
                 :
                 : "v"(lds_off), "v"(gsrc)
                 : "memory");
#else
    (void)gsrc; (void)lds_dst;
#endif
}

__device__ __forceinline__ void wait_async_zero() {
#if defined(__HIP_DEVICE_COMPILE__) && __has_builtin(__builtin_amdgcn_s_wait_asynccnt)
    __builtin_amdgcn_s_wait_asynccnt(0);
#elif defined(__HIP_DEVICE_COMPILE__)
    asm volatile("s_wait_asynccnt 0x0" ::: "memory");
#endif
}

// ---- Kernel ----------------------------------------------------------------

__global__ __launch_bounds__(BLOCK) void lif_spike_kernel(
    const float* __restrict__ x, float* __restrict__ out, int n_per_t) {
    __shared__ v4f tile[T_STEPS * BLOCK];   // 32 KB: all 8 timesteps staged

    const int tid = threadIdx.x;
    const long long base =
        (long long)blockIdx.x * CHUNK + (long long)tid * VEC;
    if (base >= (long long)n_per_t) return;   // exact division in practice

    // Fire all 8 timestep loads into LDS; one wave has 8 async b128 in flight.
    #pragma unroll
    for (int t = 0; t < T_STEPS; ++t) {
        async_load_b128(x + (long long)t * n_per_t + base,
                        (void*)&tile[t * BLOCK + tid]);
    }
    wait_async_zero();   // each thread reads only the LDS slots it requested

    float mem[VEC];
    #pragma unroll
    for (int i = 0; i < VEC; ++i) mem[i] = 0.0f;

    #pragma unroll
    for (int t = 0; t < T_STEPS; ++t) {
        v4f xv = tile[t * BLOCK + tid];      // ds_load_b128, conflict-free
        v4f spk;
        #pragma unroll
        for (int i = 0; i < VEC; ++i) {
            float m = mem[i] * BETA + xv[i];     // v_fma
            bool fired = (m >= THRESH);          // v_cmp
            spk[i] = fired ? 1.0f : 0.0f;        // v_cndmask
            mem[i] = fired ? 0.0f : m;           // v_cndmask (hard reset)
        }
        // Write-once stream: nontemporal 128-bit store.
        __builtin_nontemporal_store(
            spk, (v4f*)(out + (long long)t * n_per_t + base));
    }
}

// ---- Launch ----------------------------------------------------------------

extern "C" void kernel_launch(void* const* d_in, const int* in_sizes, int n_in,
                              void* d_out, int out_size, void* d_ws, size_t ws_size,
                              hipStream_t stream) {
    (void)n_in; (void)out_size; (void)d_ws; (void)ws_size;
    const float* x = (const float*)d_in[0];
    float* out = (float*)d_out;

    const long long n_total = (long long)in_sizes[0];      // 33,554,432
    const int n_per_t = (int)(n_total / T_STEPS);          // 4,194,304
    const int grid = (n_per_t + CHUNK - 1) / CHUNK;        // 4096

    lif_spike_kernel<<<grid, BLOCK, 0, stream>>>(x, out, n_per_t);
}